// TripletMarginWithComplexDistanceLoss_9337258901554
// MI455X (gfx1250) — compile-verified
//
#include <hip/hip_runtime.h>

// ---------------------------------------------------------------------------
// TripletMarginWithComplexDistanceLoss for MI455X (gfx1250, wave32)
//
// Memory-bound streaming problem: 805 MB read @ 23.3 TB/s => ~35us floor.
// Main kernel: coalesced non-temporal b128 streaming + wave32 shuffle reduce.
// Final reduction: V_WMMA_F32_16X16X4_F32 (B = ones) for exact f32 row-sums,
// with an unguarded b64 A-operand load (no divergence around the WMMA).
// ---------------------------------------------------------------------------

typedef float f4  __attribute__((ext_vector_type(4)));
typedef float v2f __attribute__((ext_vector_type(2)));
typedef float v8f __attribute__((ext_vector_type(8)));

#define DDIM    256
#define EPSF    1e-8f
#define INV_PI  0.31830988618379067154f
#define MARGIN  1.0f

// ---- per-element accumulation: 7 reductions, all as FMAs ------------------
__device__ __forceinline__ void acc1(float ar, float ai, float pr, float pi,
                                     float nr, float ni,
                                     float& dpr, float& dpi, float& dnr, float& dni,
                                     float& na,  float& np,  float& nn) {
  dpr = fmaf(ar, pr, fmaf(ai,  pi, dpr));   // Re(a . conj(p))
  dpi = fmaf(ai, pr, fmaf(-ar, pi, dpi));   // Im(a . conj(p))
  dnr = fmaf(ar, nr, fmaf(ai,  ni, dnr));   // Re(a . conj(n))
  dni = fmaf(ai, nr, fmaf(-ar, ni, dni));   // Im(a . conj(n))
  na  = fmaf(ar, ar, fmaf(ai, ai, na));     // |a|^2
  np  = fmaf(pr, pr, fmaf(pi, pi, np));     // |p|^2
  nn  = fmaf(nr, nr, fmaf(ni, ni, nn));     // |n|^2
}

__device__ __forceinline__ void acc4(f4 ar, f4 ai, f4 pr, f4 pi, f4 nr, f4 ni,
                                     float& dpr, float& dpi, float& dnr, float& dni,
                                     float& na,  float& np,  float& nn) {
  acc1(ar.x, ai.x, pr.x, pi.x, nr.x, ni.x, dpr, dpi, dnr, dni, na, np, nn);
  acc1(ar.y, ai.y, pr.y, pi.y, nr.y, ni.y, dpr, dpi, dnr, dni, na, np, nn);
  acc1(ar.z, ai.z, pr.z, pi.z, nr.z, ni.z, dpr, dpi, dnr, dni, na, np, nn);
  acc1(ar.w, ai.w, pr.w, pi.w, nr.w, ni.w, dpr, dpi, dnr, dni, na, np, nn);
}

// ---- principal complex arccos, scaled by 1/pi -----------------------------
// acos(z) = arg(t) - i*ln|t|,  t = z + i*sqrt(1 - z^2)  (principal sqrt).
// b = -2xy carries the signed zero, reproducing NumPy/C99 branch cuts.
__device__ __forceinline__ void cacos_over_pi(float x, float y,
                                              float& re, float& im) {
  float a  = 1.0f - x * x + y * y;       // Re(1 - z^2)
  float b  = -2.0f * x * y;              // Im(1 - z^2)
  float r  = hypotf(a, b);
  float sr = sqrtf(fmaxf(0.5f * (r + a), 0.0f));               // Re sqrt
  float si = copysignf(sqrtf(fmaxf(0.5f * (r - a), 0.0f)), b); // Im sqrt
  float tre = x - si;                    // Re(z + i*w)
  float tim = y + sr;                    // Im(z + i*w)
  re = atan2f(tim, tre) * INV_PI;
  im = -0.5f * logf(fmaf(tre, tre, tim * tim)) * INV_PI;
}

// ---- main streaming kernel: one wave32 per row, 8 rows per block ----------
__global__ __launch_bounds__(256) void triplet_rows_kernel(
    const float* __restrict__ gar, const float* __restrict__ gai,
    const float* __restrict__ gpr, const float* __restrict__ gpi,
    const float* __restrict__ gnr, const float* __restrict__ gni,
    float* __restrict__ partial, int nrows) {
  __shared__ float sdata[8];
  const int lane = threadIdx.x & 31;
  const int warp = threadIdx.x >> 5;
  const int row  = blockIdx.x * 8 + warp;

  float loss = 0.0f;
  if (row < nrows) {
    const f4* Ar = (const f4*)gar; const f4* Ai = (const f4*)gai;
    const f4* Pr = (const f4*)gpr; const f4* Pi = (const f4*)gpi;
    const f4* Nr = (const f4*)gnr; const f4* Ni = (const f4*)gni;
    const int b  = row * (DDIM / 4);        // 64 float4 per row
    const int i0 = b + lane;                // lanes cover 512B contiguous
    const int i1 = i0 + 32;

    // Streamed once, >> L2 (805MB vs 192MB): non-temporal loads.
    f4 ar0 = __builtin_nontemporal_load(Ar + i0);
    f4 ar1 = __builtin_nontemporal_load(Ar + i1);
    f4 ai0 = __builtin_nontemporal_load(Ai + i0);
    f4 ai1 = __builtin_nontemporal_load(Ai + i1);
    f4 pr0 = __builtin_nontemporal_load(Pr + i0);
    f4 pr1 = __builtin_nontemporal_load(Pr + i1);
    f4 pi0 = __builtin_nontemporal_load(Pi + i0);
    f4 pi1 = __builtin_nontemporal_load(Pi + i1);
    f4 nr0 = __builtin_nontemporal_load(Nr + i0);
    f4 nr1 = __builtin_nontemporal_load(Nr + i1);
    f4 ni0 = __builtin_nontemporal_load(Ni + i0);
    f4 ni1 = __builtin_nontemporal_load(Ni + i1);

    float dpr = 0.f, dpi = 0.f, dnr = 0.f, dni = 0.f;
    float na = 0.f, np = 0.f, nn = 0.f;
    acc4(ar0, ai0, pr0, pi0, nr0, ni0, dpr, dpi, dnr, dni, na, np, nn);
    acc4(ar1, ai1, pr1, pi1, nr1, ni1, dpr, dpi, dnr, dni, na, np, nn);

    // wave32 butterfly reduction of the 7 partials
    #pragma unroll
    for (int m = 16; m >= 1; m >>= 1) {
      dpr += __shfl_xor(dpr, m, 32);
      dpi += __shfl_xor(dpi, m, 32);
      dnr += __shfl_xor(dnr, m, 32);
      dni += __shfl_xor(dni, m, 32);
      na  += __shfl_xor(na,  m, 32);
      np  += __shfl_xor(np,  m, 32);
      nn  += __shfl_xor(nn,  m, 32);
    }

    // per-row scalar math (uniform across the wave)
    float m1  = fmaxf(sqrtf(na), EPSF);
    float m2p = fmaxf(sqrtf(np), EPSF);
    float m2n = fmaxf(sqrtf(nn), EPSF);
    float sp  = 1.0f / (m1 * m2p);
    float sn  = 1.0f / (m1 * m2n);
    float pre, pim, nre, nim;
    cacos_over_pi(dpr * sp, dpi * sp, pre, pim);
    cacos_over_pi(dnr * sn, dni * sn, nre, nim);
    float lr = fmaxf(MARGIN + pre - nre, 0.0f);
    float li = fmaxf(MARGIN + pim - nim, 0.0f);
    loss = sqrtf(fmaf(lr, lr, li * li));
  }

  if (lane == 0) sdata[warp] = loss;
  __syncthreads();
  if (threadIdx.x == 0) {
    float s = 0.0f;
    #pragma unroll
    for (int w = 0; w < 8; ++w) s += sdata[w];  // fixed order: deterministic
    partial[blockIdx.x] = s;
  }
}

// ---- final reduction via V_WMMA_F32_16X16X4_F32 ---------------------------
// A = 16x4 tile of partials, B = all-ones => every D column holds exact f32
// row-sums (independent of A/B lane striping). One wave32, EXEC all ones.
// Bulk loop is unguarded: i0 = c + 4m + kb is even => one aligned b64 load
// per iteration, zero divergence around the WMMA. Scalar tail handles P%64
// (empty for the actual shape: P = 16384).
__global__ __launch_bounds__(32) void reduce_wmma_kernel(
    const float* __restrict__ x, float* __restrict__ out, int P, float invN) {
  const int lane = threadIdx.x;
  const int m    = lane & 15;
  const int kb   = (lane >> 4) << 1;   // K pair selected by half-wave
  const int Pfull = P & ~63;

  v8f acc = {};
  v2f ones;
  ones.x = 1.0f;
  ones.y = 1.0f;

  for (int c = 0; c < Pfull; c += 64) {
    const v2f a = *(const v2f*)(x + c + m * 4 + kb);  // global_load_b64
    // (neg_a, A, neg_b, B, c_mod, C, reuse_a, reuse_b)
    acc = __builtin_amdgcn_wmma_f32_16x16x4_f32(false, a, false, ones,
                                                (short)0, acc, false, false);
  }

  // lanes 0-15 hold row-sums M=0..7 across acc[0..7]; lanes 16-31 hold M=8..15
  float s = acc[0] + acc[1] + acc[2] + acc[3] +
            acc[4] + acc[5] + acc[6] + acc[7];
  s += __shfl_xor(s, 16, 32);          // every lane: full WMMA total

  // remainder (P % 64 elements), lane-striped then fully butterflied
  float t = 0.0f;
  for (int i = Pfull + lane; i < P; i += 32) t += x[i];
  #pragma unroll
  for (int mm = 16; mm >= 1; mm >>= 1) t += __shfl_xor(t, mm, 32);

  if (lane == 0) out[0] = (s + t) * invN;
}

// ---------------------------------------------------------------------------
extern "C" void kernel_launch(void* const* d_in, const int* in_sizes, int n_in,
                              void* d_out, int out_size, void* d_ws, size_t ws_size,
                              hipStream_t stream) {
  const float* ar = (const float*)d_in[0];
  const float* ai = (const float*)d_in[1];
  const float* pr = (const float*)d_in[2];
  const float* pi = (const float*)d_in[3];
  const float* nr = (const float*)d_in[4];
  const float* ni = (const float*)d_in[5];

  const int nrows  = in_sizes[0] / DDIM;         // 131072
  const int blocks = (nrows + 7) / 8;            // 8 rows (8 waves) per block
  float* partial   = (float*)d_ws;               // blocks * 4 bytes

  triplet_rows_kernel<<<blocks, 256, 0, stream>>>(ar, ai, pr, pi, nr, ni,
                                                  partial, nrows);
  reduce_wmma_kernel<<<1, 32, 0, stream>>>(partial, (float*)d_out, blocks,
                                           1.0f / (float)nrows);
}